// SuperpixelSAN_54838142435868
// MI455X (gfx1250) — compile-verified
//
#include <hip/hip_runtime.h>
#include <cstdint>
#include <cstddef>

typedef float v2f __attribute__((ext_vector_type(2)));
typedef float v8f __attribute__((ext_vector_type(8)));

#define FSZ    18
#define OUTF   10
#define NBATCH 512
#define XCOLS  54   // concat(x1,x2,x3)

static inline int cdiv(long a, int b) { return (int)((a + b - 1) / b); }

// ---------------------------------------------------------------- utilities
__global__ void zero_f32(float* __restrict__ p, long n) {
  long i = (long)blockIdx.x * blockDim.x + threadIdx.x;
  if (i < n) p[i] = 0.f;
}

// monotone float<->uint encoding so atomicMax(uint) == float max
__device__ __forceinline__ unsigned enc_f32(float f) {
  unsigned u = __float_as_uint(f);
  return u ^ ((u >> 31) ? 0xFFFFFFFFu : 0x80000000u);
}
__device__ __forceinline__ float dec_f32(unsigned u) {
  u ^= ((u >> 31) ? 0x80000000u : 0xFFFFFFFFu);
  return __uint_as_float(u);
}

// ------------------------------------------------- fp32 WMMA dense: Y = X@W + b
// X: [n, *] row-stride ldx, use K columns.  W: [K,F] row-major.  Y: [n,F].
// One wave -> 16x32 output tile (two 16x16 f32 WMMA accumulators), 4 waves/block.
// W is staged to LDS column-major (stride 66 floats -> 8B-aligned b64 reads,
// bank index (2c + k) % 64) so every B fragment is a single ds_load_b64.
__global__ void __launch_bounds__(128)
gemm_bias_wmma(const float* __restrict__ X, int ldx, int K,
               const float* __restrict__ W, const float* __restrict__ bias,
               float* __restrict__ Y, int F, int n) {
  __shared__ float sW[32 * 66];
  __shared__ float sB[32];
  for (int idx = threadIdx.x; idx < 32 * 64; idx += blockDim.x) {
    int c = idx >> 6, kk = idx & 63;
    sW[c * 66 + kk] = (kk < K && c < F) ? W[kk * F + c] : 0.f;
  }
  if (threadIdx.x < 32) sB[threadIdx.x] = (threadIdx.x < F) ? bias[threadIdx.x] : 0.f;
  __syncthreads();

  const int lane  = threadIdx.x & 31;
  const int wave  = threadIdx.x >> 5;
  const int row0  = (blockIdx.x * 4 + wave) * 16;
  const int col   = lane & 15;
  const int khalf = (lane >> 4) << 1;             // lanes 0-15 -> K {k,k+1}; 16-31 -> {k+2,k+3}
  int r  = row0 + col;
  int rc = (r < n) ? r : (n - 1);                 // clamp for safe loads; stores are guarded
  const float* xbase = X + (size_t)rc * ldx;
  const float* xrow  = xbase + khalf;
  const float* wcol0 = sW + col * 66 + khalf;
  const float* wcol1 = sW + (col + 16) * 66 + khalf;

  v8f acc0 = {0.f,0.f,0.f,0.f,0.f,0.f,0.f,0.f};
  v8f acc1 = {0.f,0.f,0.f,0.f,0.f,0.f,0.f,0.f};

  const int Kmain = K & ~3;                       // unguarded full quads
  int k = 0;
  #pragma unroll 4
  for (; k < Kmain; k += 4) {
    v2f a  = *(const v2f*)(xrow + k);             // global_load_b64
    v2f b0 = *(const v2f*)(wcol0 + k);            // ds_load_b64
    v2f b1 = *(const v2f*)(wcol1 + k);            // ds_load_b64
    acc0 = __builtin_amdgcn_wmma_f32_16x16x4_f32(false, a, false, b0, (short)0, acc0, false, false);
    acc1 = __builtin_amdgcn_wmma_f32_16x16x4_f32(false, a, false, b1, (short)0, acc1, false, false);
  }
  if (K & 3) {                                    // one padded tail quad (K = 18 / 54)
    int kk = k + khalf;
    int c0 = (kk     < K) ? kk     : K - 1;       // clamped unconditional loads,
    int c1 = (kk + 1 < K) ? kk + 1 : K - 1;       // masked with cndmask (no exec churn)
    float ax = xbase[c0];
    float ay = xbase[c1];
    v2f a;
    a.x = (kk     < K) ? ax : 0.f;
    a.y = (kk + 1 < K) ? ay : 0.f;
    v2f b0 = *(const v2f*)(wcol0 + k);            // LDS is zero-padded past K
    v2f b1 = *(const v2f*)(wcol1 + k);
    acc0 = __builtin_amdgcn_wmma_f32_16x16x4_f32(false, a, false, b0, (short)0, acc0, false, false);
    acc1 = __builtin_amdgcn_wmma_f32_16x16x4_f32(false, a, false, b1, (short)0, acc1, false, false);
  }

  const int mofs = (lane >> 4) << 3;              // lanes 16-31 hold M = 8..15
  #pragma unroll
  for (int i = 0; i < 8; ++i) {
    int rr = row0 + mofs + i;
    if (rr < n) {
      if (col < F)      Y[(size_t)rr * F + col]      = acc0[i] + sB[col];
      if (col + 16 < F) Y[(size_t)rr * F + col + 16] = acc1[i] + sB[col + 16];
    }
  }
}

// ------------------------------------------------- SpMM: acc[dst] += v * H[src]
__global__ void spmm_edges(const int* __restrict__ dst, const int* __restrict__ src,
                           const float* __restrict__ vals, const float* __restrict__ H,
                           float* __restrict__ acc, long E) {
  long e = (long)blockIdx.x * blockDim.x + threadIdx.x;
  if (e >= E) return;
  int d = dst[e], s = src[e];
  float v = vals[e];
  const v2f* h = (const v2f*)(H + (size_t)s * FSZ);   // rows are 72B -> 8B aligned
  float* o = acc + (size_t)d * FSZ;
  #pragma unroll
  for (int f = 0; f < FSZ / 2; ++f) {
    v2f hv = h[f];
    atomicAdd(&o[2 * f],     v * hv.x);
    atomicAdd(&o[2 * f + 1], v * hv.y);
  }
}

// ------------------------------------------------- GAT pieces
__global__ void gat_scores(const float* __restrict__ H,
                           const float* __restrict__ a_src, const float* __restrict__ a_dst,
                           float* __restrict__ ssrc, float* __restrict__ sdst, int n) {
  int i = blockIdx.x * blockDim.x + threadIdx.x;
  if (i >= n) return;
  const v2f* h  = (const v2f*)(H + (size_t)i * FSZ);
  const v2f* as = (const v2f*)a_src;
  const v2f* ad = (const v2f*)a_dst;
  float s0 = 0.f, s1 = 0.f;
  #pragma unroll
  for (int f = 0; f < FSZ / 2; ++f) {
    v2f hv = h[f], av = as[f], dv = ad[f];
    s0 += hv.x * av.x + hv.y * av.y;
    s1 += hv.x * dv.x + hv.y * dv.y;
  }
  ssrc[i] = s0; sdst[i] = s1;
}

__global__ void gat_edge_max(const int* __restrict__ dst, const int* __restrict__ src,
                             const float* __restrict__ ssrc, const float* __restrict__ sdst,
                             unsigned* __restrict__ menc, long E) {
  long e = (long)blockIdx.x * blockDim.x + threadIdx.x;
  if (e >= E) return;
  int d = dst[e];
  float v = ssrc[src[e]] + sdst[d];
  v = (v > 0.f) ? v : 0.2f * v;                   // leaky_relu(.,0.2)
  atomicMax(&menc[d], enc_f32(v));
}

__global__ void gat_edge_scatter(const int* __restrict__ dst, const int* __restrict__ src,
                                 const float* __restrict__ ssrc, const float* __restrict__ sdst,
                                 const unsigned* __restrict__ menc, const float* __restrict__ H,
                                 float* __restrict__ num, float* __restrict__ den, long E) {
  long e = (long)blockIdx.x * blockDim.x + threadIdx.x;
  if (e >= E) return;
  int d = dst[e], s = src[e];
  float v = ssrc[s] + sdst[d];
  v = (v > 0.f) ? v : 0.2f * v;
  unsigned me = menc[d];
  float m = me ? dec_f32(me) : 0.f;               // empty-segment sentinel -> 0 (matches ref)
  float w = expf(v - m);
  atomicAdd(&den[d], w);
  const v2f* h = (const v2f*)(H + (size_t)s * FSZ);
  float* o = num + (size_t)d * FSZ;
  #pragma unroll
  for (int f = 0; f < FSZ / 2; ++f) {
    v2f hv = h[f];
    atomicAdd(&o[2 * f],     w * hv.x);
    atomicAdd(&o[2 * f + 1], w * hv.y);
  }
}

__global__ void gat_combine(const float* __restrict__ num, const float* __restrict__ den,
                            float* __restrict__ acc, int n) {
  int i = blockIdx.x * blockDim.x + threadIdx.x;
  if (i >= n) return;
  float inv = 1.f / fmaxf(den[i], 1e-9f);
  const v2f* nu = (const v2f*)(num + (size_t)i * FSZ);
  v2f* ac = (v2f*)(acc + (size_t)i * FSZ);
  #pragma unroll
  for (int f = 0; f < FSZ / 2; ++f) {
    v2f a = ac[f], x = nu[f];
    a.x += x.x * inv; a.y += x.y * inv;
    ac[f] = a;
  }
}

__global__ void relu_store(const float* __restrict__ acc, float* __restrict__ X123,
                           int col0, int n) {
  int i = blockIdx.x * blockDim.x + threadIdx.x;
  if (i >= n) return;
  const v2f* a = (const v2f*)(acc + (size_t)i * FSZ);
  v2f* xo = (v2f*)(X123 + (size_t)i * XCOLS + col0);  // col0 in {0,18,36} -> 8B aligned
  #pragma unroll
  for (int f = 0; f < FSZ / 2; ++f) {
    v2f v = a[f];
    v.x = fmaxf(v.x, 0.f); v.y = fmaxf(v.y, 0.f);
    xo[f] = v;
  }
}

// ------------------------------------------------- pooling + head
__global__ void pool_accum(const float* __restrict__ X4, const int* __restrict__ batch,
                           float* __restrict__ sums, float* __restrict__ cnt, int n) {
  int i = blockIdx.x * blockDim.x + threadIdx.x;
  if (i >= n) return;
  int b = batch[i];
  atomicAdd(&cnt[b], 1.f);
  const v2f* x = (const v2f*)(X4 + (size_t)i * OUTF);  // 40B rows -> 8B aligned
  float* o = sums + (size_t)b * OUTF;
  #pragma unroll
  for (int f = 0; f < OUTF / 2; ++f) {
    v2f v = x[f];
    atomicAdd(&o[2 * f],     v.x);
    atomicAdd(&o[2 * f + 1], v.y);
  }
}

__global__ void pool_div(const float* __restrict__ sums, const float* __restrict__ cnt,
                         float* __restrict__ pooled, int colofs, int nb) {
  int b = blockIdx.x * blockDim.x + threadIdx.x;
  if (b >= nb) return;
  float c = fmaxf(cnt[b], 1.f);
  #pragma unroll
  for (int f = 0; f < OUTF; ++f)
    pooled[(size_t)b * (3 * OUTF) + colofs + f] = sums[(size_t)b * OUTF + f] / c;
}

__global__ void final_head(const float* __restrict__ pooled, const float* __restrict__ Wcomb,
                           const float* __restrict__ bcomb, float* __restrict__ out, int nb) {
  int b = blockIdx.x * blockDim.x + threadIdx.x;
  if (b >= nb) return;
  float lg[OUTF];
  #pragma unroll
  for (int j = 0; j < OUTF; ++j) lg[j] = bcomb[j];
  for (int k = 0; k < 3 * OUTF; ++k) {
    float p = pooled[(size_t)b * (3 * OUTF) + k];
    #pragma unroll
    for (int j = 0; j < OUTF; ++j) lg[j] += p * Wcomb[k * OUTF + j];
  }
  float mx = lg[0];
  #pragma unroll
  for (int j = 1; j < OUTF; ++j) mx = fmaxf(mx, lg[j]);
  float s = 0.f;
  #pragma unroll
  for (int j = 0; j < OUTF; ++j) { lg[j] = expf(lg[j] - mx); s += lg[j]; }
  float inv = 1.f / s;
  #pragma unroll
  for (int j = 0; j < OUTF; ++j) out[(size_t)b * OUTF + j] = lg[j] * inv;
}

// ================================================= host orchestration
namespace {

struct Ws {
  float* X123; float* Hb; float* acc; float* num; float* den;
  unsigned* menc; float* ssrc; float* sdst; float* X4;
  float* sums; float* cnt; float* pooled;
};

void gat_branch(hipStream_t st, int n, const float* Xin, int ldx, int K,
                const float* W, const float* bvec,
                const float* a_s, const float* a_d,
                const int* dst, const int* src, long E, const Ws& w) {
  zero_f32<<<cdiv((long)n * FSZ, 256), 256, 0, st>>>(w.num, (long)n * FSZ);
  zero_f32<<<cdiv(n, 256), 256, 0, st>>>(w.den, n);
  zero_f32<<<cdiv(n, 256), 256, 0, st>>>((float*)w.menc, n);
  gemm_bias_wmma<<<cdiv(n, 64), 128, 0, st>>>(Xin, ldx, K, W, bvec, w.Hb, FSZ, n);
  gat_scores<<<cdiv(n, 256), 256, 0, st>>>(w.Hb, a_s, a_d, w.ssrc, w.sdst, n);
  gat_edge_max<<<cdiv(E, 256), 256, 0, st>>>(dst, src, w.ssrc, w.sdst, w.menc, E);
  gat_edge_scatter<<<cdiv(E, 256), 256, 0, st>>>(dst, src, w.ssrc, w.sdst, w.menc,
                                                 w.Hb, w.num, w.den, E);
  gat_combine<<<cdiv(n, 256), 256, 0, st>>>(w.num, w.den, w.acc, n);
}

void san_layer(hipStream_t st, int n, const float* Xin, int ldx, int K,
               const float* Wbase, const float* bbase, size_t wstride, size_t bstride,
               const float* a_u_s, const float* a_u_d,
               const float* a_d_s, const float* a_d_d,
               const int* ep_dst, const int* ep_src, const float* pvals, long Ep,
               bool has_u, const int* eu_dst, const int* eu_src, long Eu,
               bool has_d, const int* ed_dst, const int* ed_src, long Ed,
               int outcol, const Ws& w) {
  zero_f32<<<cdiv((long)n * FSZ, 256), 256, 0, st>>>(w.acc, (long)n * FSZ);
  // p branch (branch index 2)
  gemm_bias_wmma<<<cdiv(n, 64), 128, 0, st>>>(Xin, ldx, K, Wbase + 2 * wstride,
                                              bbase + 2 * bstride, w.Hb, FSZ, n);
  spmm_edges<<<cdiv(Ep, 256), 256, 0, st>>>(ep_dst, ep_src, pvals, w.Hb, w.acc, Ep);
  if (has_u)
    gat_branch(st, n, Xin, ldx, K, Wbase + 0 * wstride, bbase + 0 * bstride,
               a_u_s, a_u_d, eu_dst, eu_src, Eu, w);
  if (has_d)
    gat_branch(st, n, Xin, ldx, K, Wbase + 1 * wstride, bbase + 1 * bstride,
               a_d_s, a_d_d, ed_dst, ed_src, Ed, w);
  relu_store<<<cdiv(n, 256), 256, 0, st>>>(w.acc, w.X123, outcol, n);
}

void run_level(hipStream_t st, int L, int n, const float* X,
               const int* ep, const float* pvals, long Ep,
               bool has_u, const int* eu, long Eu,
               bool has_d, const int* ed, long Ed,
               const float* W1a, const float* b1a, const float* a1a,
               const float* W23a, const float* b23a, const float* a23a,
               const float* Wouta, const float* bouta,
               const int* batch, const Ws& w) {
  const int DIN = 64;
  const float* W1  = W1a  + (size_t)L * 3 * DIN * FSZ;
  const float* b1  = b1a  + (size_t)L * 3 * FSZ;
  const float* a1  = a1a  + (size_t)L * 2 * 2 * FSZ;
  const float* W23 = W23a + (size_t)L * 2 * 3 * FSZ * FSZ;
  const float* b23 = b23a + (size_t)L * 2 * 3 * FSZ;
  const float* a23 = a23a + (size_t)L * 2 * 2 * 2 * FSZ;
  const float* Wout = Wouta + (size_t)L * XCOLS * OUTF;
  const float* bout = bouta + (size_t)L * OUTF;

  const int* ep_d = ep;      const int* ep_s = ep + Ep;
  const int* eu_d = eu;      const int* eu_s = has_u ? eu + Eu : nullptr;
  const int* ed_d = ed;      const int* ed_s = has_d ? ed + Ed : nullptr;

  // layer 1: DIN -> F
  san_layer(st, n, X, DIN, DIN, W1, b1, (size_t)DIN * FSZ, FSZ,
            a1 + 0 * FSZ, a1 + 1 * FSZ, a1 + 2 * FSZ, a1 + 3 * FSZ,
            ep_d, ep_s, pvals, Ep, has_u, eu_d, eu_s, Eu, has_d, ed_d, ed_s, Ed,
            0, w);
  // layers 2,3: F -> F, input = previous column slice of X123
  for (int l = 0; l < 2; ++l) {
    const float* Wl = W23 + (size_t)l * 3 * FSZ * FSZ;
    const float* bl = b23 + (size_t)l * 3 * FSZ;
    const float* al = a23 + (size_t)l * 2 * 2 * FSZ;
    san_layer(st, n, w.X123 + (size_t)l * FSZ, XCOLS, FSZ, Wl, bl,
              (size_t)FSZ * FSZ, FSZ,
              al + 0 * FSZ, al + 1 * FSZ, al + 2 * FSZ, al + 3 * FSZ,
              ep_d, ep_s, pvals, Ep, has_u, eu_d, eu_s, Eu, has_d, ed_d, ed_s, Ed,
              (l + 1) * FSZ, w);
  }
  // readout 54 -> 10 and mean pool
  gemm_bias_wmma<<<cdiv(n, 64), 128, 0, st>>>(w.X123, XCOLS, XCOLS, Wout, bout,
                                              w.X4, OUTF, n);
  zero_f32<<<cdiv(NBATCH * OUTF, 256), 256, 0, st>>>(w.sums, NBATCH * OUTF);
  zero_f32<<<cdiv(NBATCH, 256), 256, 0, st>>>(w.cnt, NBATCH);
  pool_accum<<<cdiv(n, 256), 256, 0, st>>>(w.X4, batch, w.sums, w.cnt, n);
  pool_div<<<cdiv(NBATCH, 128), 128, 0, st>>>(w.sums, w.cnt, w.pooled, L * OUTF, NBATCH);
}

} // namespace

extern "C" void kernel_launch(void* const* d_in, const int* in_sizes, int n_in,
                              void* d_out, int out_size, void* d_ws, size_t ws_size,
                              hipStream_t stream) {
  const float* W1    = (const float*)d_in[0];
  const float* b1    = (const float*)d_in[1];
  const float* a1    = (const float*)d_in[2];
  const float* W23   = (const float*)d_in[3];
  const float* b23   = (const float*)d_in[4];
  const float* a23   = (const float*)d_in[5];
  const float* Wout  = (const float*)d_in[6];
  const float* bout  = (const float*)d_in[7];
  const float* Wcomb = (const float*)d_in[8];
  const float* bcomb = (const float*)d_in[9];
  const float* vL0   = (const float*)d_in[10];
  const float* vL1   = (const float*)d_in[11];
  const float* vL2   = (const float*)d_in[14];
  const float* X0    = (const float*)d_in[15];
  const float* X1    = (const float*)d_in[16];
  const float* X2    = (const float*)d_in[17];
  const int*   eL0   = (const int*)d_in[18];
  const int*   eL1   = (const int*)d_in[19];
  const int*   eL1u  = (const int*)d_in[20];
  const int*   eL1d  = (const int*)d_in[21];
  const int*   eL2   = (const int*)d_in[22];
  const int*   bt0   = (const int*)d_in[23];
  const int*   bt1   = (const int*)d_in[24];
  const int*   bt2   = (const int*)d_in[25];

  const long E0 = in_sizes[10], E1 = in_sizes[11], EU = in_sizes[12],
             ED = in_sizes[13], E2 = in_sizes[14];
  const int n0 = in_sizes[15] / 64, n1 = in_sizes[16] / 64, n2 = in_sizes[17] / 64;
  const size_t NMAX = (size_t)((n0 > n1 ? (n0 > n2 ? n0 : n2) : (n1 > n2 ? n1 : n2)));

  float* base = (float*)d_ws;
  size_t o = 0;
  auto carve = [&](size_t cnt) { float* p = base + o; o += (cnt + 63) & ~(size_t)63; return p; };
  Ws w;
  w.X123  = carve(NMAX * XCOLS);
  w.Hb    = carve(NMAX * FSZ);
  w.acc   = carve(NMAX * FSZ);
  w.num   = carve(NMAX * FSZ);
  w.den   = carve(NMAX);
  w.menc  = (unsigned*)carve(NMAX);
  w.ssrc  = carve(NMAX);
  w.sdst  = carve(NMAX);
  w.X4    = carve(NMAX * OUTF);
  w.sums  = carve((size_t)NBATCH * OUTF);
  w.cnt   = carve((size_t)NBATCH);
  w.pooled= carve((size_t)NBATCH * 3 * OUTF);

  // level 0: p over eL0, d-GAT over eL0 (no u)
  run_level(stream, 0, n0, X0, eL0, vL0, E0,
            /*has_u*/false, nullptr, 0, /*has_d*/true, eL0, E0,
            W1, b1, a1, W23, b23, a23, Wout, bout, bt0, w);
  // level 1: p over eL1, u-GAT over eL1u, d-GAT over eL1d
  run_level(stream, 1, n1, X1, eL1, vL1, E1,
            true, eL1u, EU, true, eL1d, ED,
            W1, b1, a1, W23, b23, a23, Wout, bout, bt1, w);
  // level 2: p over eL2, u-GAT over eL2 (no d)
  run_level(stream, 2, n2, X2, eL2, vL2, E2,
            true, eL2, E2, false, nullptr, 0,
            W1, b1, a1, W23, b23, a23, Wout, bout, bt2, w);

  final_head<<<cdiv(NBATCH, 128), 128, 0, stream>>>(w.pooled, Wcomb, bcomb,
                                                    (float*)d_out, NBATCH);
}